// LogEig_15839839388187
// MI455X (gfx1250) — compile-verified
//
#include <hip/hip_runtime.h>
#include <math.h>

// LogEig: batched 64x64 SPD eigendecomposition (parallel cyclic Jacobi) +
// V*diag(log w)*V^T reconstruction via V_WMMA_F32_16X16X4_F32.
//
// One workgroup (256 threads = 8 wave32) per matrix. A and V live in LDS with
// row stride 65 (bank-conflict-free for both row and column access patterns).

typedef float v2f __attribute__((ext_vector_type(2)));
typedef float v8f __attribute__((ext_vector_type(8)));

#define NN       64
#define LDAP     65          // LDS row stride (floats): (65*i + p) % 64 conflict-free
#define NPAIR    32
#define NROUNDS  63
#define NSWEEPS  9
#define TPB      256

__launch_bounds__(TPB)
__global__ void logeig_kernel(const float* __restrict__ X, float* __restrict__ out)
{
    __shared__ float As[NN * LDAP];   // working matrix, later reused as W = V*diag(logw)
    __shared__ float Vs[NN * LDAP];   // accumulated eigenvectors
    __shared__ float cs[NPAIR];
    __shared__ float sn[NPAIR];
    __shared__ int   pq[NPAIR];       // p | (q<<8)
    __shared__ float lw[NN];          // log eigenvalues

    const int tid = threadIdx.x;
    const long long base = (long long)blockIdx.x * (NN * NN);

    // ---- Load X (coalesced b128) and init V = I ----------------------------
    #pragma unroll
    for (int it = 0; it < 4; ++it) {
        const int e   = (it * TPB + tid) * 4;          // 0..4095, 4-aligned
        const int row = e >> 6;
        const int col = e & 63;
        const float4 v = *reinterpret_cast<const float4*>(X + base + e);
        float* ad = &As[row * LDAP + col];
        ad[0] = v.x; ad[1] = v.y; ad[2] = v.z; ad[3] = v.w;
        float* vd = &Vs[row * LDAP + col];
        vd[0] = (col + 0 == row) ? 1.0f : 0.0f;
        vd[1] = (col + 1 == row) ? 1.0f : 0.0f;
        vd[2] = (col + 2 == row) ? 1.0f : 0.0f;
        vd[3] = (col + 3 == row) ? 1.0f : 0.0f;
    }
    __syncthreads();

    // ---- Parallel cyclic Jacobi sweeps ------------------------------------
    for (int sweep = 0; sweep < NSWEEPS; ++sweep) {
        for (int r = 0; r < NROUNDS; ++r) {
            // 32 lanes compute the 32 disjoint rotations (round-robin schedule)
            if (tid < NPAIR) {
                const int p = (tid == 0) ? 0 : 1 + ((tid - 1 + r) % NROUNDS);
                const int q = 1 + ((NROUNDS - 1 - tid + r) % NROUNDS);
                const float app = As[p * LDAP + p];
                const float aqq = As[q * LDAP + q];
                const float apq = As[p * LDAP + q];
                float c = 1.0f, s = 0.0f;
                if (fabsf(apq) > 1e-30f) {
                    const float tau = (aqq - app) / (2.0f * apq);
                    const float t   = copysignf(1.0f, tau) /
                                      (fabsf(tau) + sqrtf(1.0f + tau * tau));
                    c = 1.0f / sqrtf(1.0f + t * t);   // smaller-angle Schur rotation
                    s = t * c;
                }
                cs[tid] = c;
                sn[tid] = s;
                pq[tid] = p | (q << 8);
            }
            __syncthreads();

            // Phase 1: column rotations, A <- A*J and V <- V*J (4096 pair-updates)
            #pragma unroll
            for (int it = 0; it < 16; ++it) {
                const int idx = it * TPB + tid;        // 0..4095
                float* M = (idx < 2048) ? As : Vs;
                const int k = (idx >> 6) & 31;
                const int i = idx & 63;
                const int e = pq[k];
                const int p = e & 255, q = e >> 8;
                const float c = cs[k], s = sn[k];
                const float x = M[i * LDAP + p];
                const float y = M[i * LDAP + q];
                M[i * LDAP + p] = c * x - s * y;
                M[i * LDAP + q] = s * x + c * y;
            }
            __syncthreads();

            // Phase 2: row rotations, A <- J^T*A (2048 pair-updates)
            #pragma unroll
            for (int it = 0; it < 8; ++it) {
                const int idx = it * TPB + tid;        // 0..2047
                const int k = idx >> 6;
                const int j = idx & 63;
                const int e = pq[k];
                const int p = e & 255, q = e >> 8;
                const float c = cs[k], s = sn[k];
                const float x = As[p * LDAP + j];
                const float y = As[q * LDAP + j];
                As[p * LDAP + j] = c * x - s * y;
                As[q * LDAP + j] = s * x + c * y;
            }
            __syncthreads();
        }
    }

    // ---- Eigenvalues: diag(A); logw = log(clamp(w, 1e-6)) -----------------
    if (tid < NN)
        lw[tid] = logf(fmaxf(As[tid * LDAP + tid], 1e-6f));
    __syncthreads();

    // ---- W = V * diag(logw), overwrite A buffer ---------------------------
    #pragma unroll
    for (int it = 0; it < 16; ++it) {
        const int idx = it * TPB + tid;
        const int i = idx >> 6, k = idx & 63;
        As[i * LDAP + k] = Vs[i * LDAP + k] * lw[k];
    }
    __syncthreads();

    // ---- out = W * V^T via V_WMMA_F32_16X16X4_F32 -------------------------
    // 4x4 grid of 16x16 output tiles; wave w owns tiles 2w and 2w+1 (same M-tile,
    // adjacent N-tiles -> A-fragment shared). K accumulated in 16 steps of 4.
    //
    // A-frag (16x4 f32, ISA 7.12.2): lane(0..15)=M, lanes>=16 hold K+2; reg r -> K+r.
    // B-frag (4x16):                 lane%16=N,   lanes>=16 hold K+2; reg r -> K+r.
    // C/D (16x16 f32):               lane%16=N, reg r -> M=r (+8 for lanes>=16).
    const int wv   = tid >> 5;
    const int lane = tid & 31;
    const int mt   = wv >> 1;                 // tile row (0..3)
    const int nt0  = (2 * wv) & 3;            // tile cols
    const int nt1  = (2 * wv + 1) & 3;
    const int krow = (lane >= 16) ? 2 : 0;
    const int aM   = mt  * 16 + (lane & 15);
    const int bN0  = nt0 * 16 + (lane & 15);
    const int bN1  = nt1 * 16 + (lane & 15);

    v8f acc0 = {};
    v8f acc1 = {};
    #pragma unroll
    for (int kk = 0; kk < 16; ++kk) {
        const int kb = kk * 4 + krow;
        v2f a, b0, b1;
        a[0]  = As[aM  * LDAP + kb];     a[1]  = As[aM  * LDAP + kb + 1];
        b0[0] = Vs[bN0 * LDAP + kb];     b0[1] = Vs[bN0 * LDAP + kb + 1];  // B[k][n]=V[n][k]
        b1[0] = Vs[bN1 * LDAP + kb];     b1[1] = Vs[bN1 * LDAP + kb + 1];
        acc0 = __builtin_amdgcn_wmma_f32_16x16x4_f32(
                   false, a, false, b0, (short)0, acc0, false, false);
        acc1 = __builtin_amdgcn_wmma_f32_16x16x4_f32(
                   false, a, false, b1, (short)0, acc1, false, false);
    }

    float* o = out + base;
    const int mrow = mt * 16 + ((lane >= 16) ? 8 : 0);
    #pragma unroll
    for (int rr = 0; rr < 8; ++rr) {
        const int M = mrow + rr;
        o[M * NN + nt0 * 16 + (lane & 15)] = acc0[rr];
        o[M * NN + nt1 * 16 + (lane & 15)] = acc1[rr];
    }
}

extern "C" void kernel_launch(void* const* d_in, const int* in_sizes, int n_in,
                              void* d_out, int out_size, void* d_ws, size_t ws_size,
                              hipStream_t stream)
{
    const float* X  = (const float*)d_in[0];
    float* out      = (float*)d_out;
    const int batch = in_sizes[0] / (NN * NN);   // 8192
    logeig_kernel<<<dim3(batch), dim3(TPB), 0, stream>>>(X, out);
}